// PointnetSAModuleMSG_34651796144290
// MI455X (gfx1250) — compile-verified
//
#include <hip/hip_runtime.h>

typedef __attribute__((ext_vector_type(2))) float v2f;
typedef __attribute__((ext_vector_type(8))) float v8f;

#define NBATCH 16
#define NPTS   4096
#define NCHAN  64
#define NPOINT 1024
#define EPSV   1e-5f

// ---------------------------------------------------------------------------
// Furthest point sampling: one block per batch, sequential over 1024 picks.
// ---------------------------------------------------------------------------
__global__ __launch_bounds__(1024)
void fps_kernel(const float* __restrict__ xyz, float* __restrict__ new_xyz) {
  __shared__ float sd[NPTS];
  __shared__ float redv[1024];
  __shared__ int   redi[1024];
  const int b   = blockIdx.x;
  const int tid = threadIdx.x;
  const float* xb = xyz + (size_t)b * NPTS * 3;

  for (int i = tid; i < NPTS; i += 1024) sd[i] = 1e10f;
  __syncthreads();

  int last = 0;
  if (tid == 0) {
    new_xyz[((size_t)b * NPOINT) * 3 + 0] = xb[0];
    new_xyz[((size_t)b * NPOINT) * 3 + 1] = xb[1];
    new_xyz[((size_t)b * NPOINT) * 3 + 2] = xb[2];
  }
  for (int it = 1; it < NPOINT; ++it) {
    float lx = xb[last * 3 + 0], ly = xb[last * 3 + 1], lz = xb[last * 3 + 2];
    float bv = -1.0f; int bi = 0;
    for (int i = tid; i < NPTS; i += 1024) {
      float dx = xb[i * 3 + 0] - lx;
      float dy = xb[i * 3 + 1] - ly;
      float dz = xb[i * 3 + 2] - lz;
      float d  = dx * dx + dy * dy + dz * dz;
      float nd = fminf(sd[i], d);
      sd[i] = nd;
      if (nd > bv) { bv = nd; bi = i; }
    }
    redv[tid] = bv; redi[tid] = bi;
    __syncthreads();
    for (int sft = 512; sft > 0; sft >>= 1) {
      if (tid < sft) {
        float v2 = redv[tid + sft]; int i2 = redi[tid + sft];
        float v1 = redv[tid];       int i1 = redi[tid];
        if (v2 > v1 || (v2 == v1 && i2 < i1)) { redv[tid] = v2; redi[tid] = i2; }
      }
      __syncthreads();
    }
    last = redi[0];
    if (tid == 0) {
      new_xyz[((size_t)b * NPOINT + it) * 3 + 0] = xb[last * 3 + 0];
      new_xyz[((size_t)b * NPOINT + it) * 3 + 1] = xb[last * 3 + 1];
      new_xyz[((size_t)b * NPOINT + it) * 3 + 2] = xb[last * 3 + 2];
    }
    __syncthreads();
  }
}

// ---------------------------------------------------------------------------
// Ball query: one wave32 per center. First K in-range indices (ascending),
// padded with the first valid index (0 if none).
// ---------------------------------------------------------------------------
__global__ __launch_bounds__(256)
void ball_query_kernel(const float* __restrict__ xyz, const float* __restrict__ new_xyz,
                       int* __restrict__ idx, float r2, int K) {
  const int gwave = (blockIdx.x * blockDim.x + threadIdx.x) >> 5;
  const int lane  = threadIdx.x & 31;
  const int b = gwave / NPOINT;
  const int s = gwave % NPOINT;
  const float cx = new_xyz[((size_t)b * NPOINT + s) * 3 + 0];
  const float cy = new_xyz[((size_t)b * NPOINT + s) * 3 + 1];
  const float cz = new_xyz[((size_t)b * NPOINT + s) * 3 + 2];
  const float* xb = xyz + (size_t)b * NPTS * 3;
  int* out = idx + ((size_t)b * NPOINT + s) * K;

  int cnt = 0, firstIdx = 0;
  for (int base = 0; base < NPTS; base += 32) {
    int j = base + lane;
    float dx = xb[j * 3 + 0] - cx;
    float dy = xb[j * 3 + 1] - cy;
    float dz = xb[j * 3 + 2] - cz;
    bool inr = (dx * dx + dy * dy + dz * dz) < r2;
    unsigned mask = (unsigned)__ballot(inr);
    if (cnt == 0 && mask) firstIdx = base + __ffs(mask) - 1;
    if (inr) {
      int pos = cnt + __popc(mask & ((1u << lane) - 1u));
      if (pos < K) out[pos] = j;
    }
    cnt += __popc(mask);
    if (cnt >= K) break;
  }
  if (cnt < K) {
    for (int p = cnt + lane; p < K; p += 32) out[p] = firstIdx;
  }
}

// ---------------------------------------------------------------------------
// Grouping: build X = (B, 3+C, S*K): centered xyz + gathered features.
// ---------------------------------------------------------------------------
__global__ __launch_bounds__(256)
void group_kernel(const float* __restrict__ xyz, const float* __restrict__ new_xyz,
                  const float* __restrict__ feat, const int* __restrict__ idx,
                  float* __restrict__ Xo, int K) {
  const size_t SK = (size_t)NPOINT * K;
  size_t t = (size_t)blockIdx.x * blockDim.x + threadIdx.x;
  const int k = (int)(t % K);
  const int s = (int)((t / K) % NPOINT);
  const int b = (int)(t / SK);
  const int j = idx[t];
  const float* xb = xyz + (size_t)b * NPTS * 3;
  const float nx = new_xyz[((size_t)b * NPOINT + s) * 3 + 0];
  const float ny = new_xyz[((size_t)b * NPOINT + s) * 3 + 1];
  const float nz = new_xyz[((size_t)b * NPOINT + s) * 3 + 2];
  float* xo = Xo + (size_t)b * (3 + NCHAN) * SK + (size_t)s * K + k;
  xo[0]          = xb[j * 3 + 0] - nx;
  xo[SK]         = xb[j * 3 + 1] - ny;
  xo[2 * SK]     = xb[j * 3 + 2] - nz;
  const float* fb = feat + (size_t)b * NCHAN * NPTS + j;
  for (int c = 0; c < NCHAN; ++c)
    xo[(size_t)(3 + c) * SK] = fb[(size_t)c * NPTS];
}

// ---------------------------------------------------------------------------
// Conv (1x1) via V_WMMA_F32_16X16X4_F32. Compile-time shapes: no integer
// division, fully unrolled K-loop. Wave tile: 32 output channels x 64 cols
// (2 A-operands share every B load -> halved X re-read redundancy).
// Fused: input BN+ReLU (prev layer) on the B-load path, and output BN stats
// (shfl_xor lane reduction + float atomics) in the epilogue.
// ---------------------------------------------------------------------------
template <int CIN, int O, int M, int RELU>
__global__ __launch_bounds__(256)
void conv_wmma_kernel(const float* __restrict__ X, const float* __restrict__ W,
                      float* __restrict__ Y,
                      const float* __restrict__ insc, const float* __restrict__ inbi,
                      float* __restrict__ osum, float* __restrict__ osumsq) {
  constexpr int MT = M / 64;    // 64-column tiles per batch
  constexpr int OT = O / 32;    // 32-row output tiles
  const int wave = (blockIdx.x * 256 + threadIdx.x) >> 5;
  const int lane = threadIdx.x & 31;
  const int half = lane >> 4;
  const int lm   = lane & 15;
  const int mt = wave % MT;
  const int ot = (wave / MT) % OT;
  const int b  = wave / (MT * OT);
  const int o0 = ot * 32;
  const int n  = mt * 64 + lm;

  const float* Xb = X + (size_t)b * CIN * M;
  const v8f zero = {0.f, 0.f, 0.f, 0.f, 0.f, 0.f, 0.f, 0.f};
  v8f acc[2][4] = {{zero, zero, zero, zero}, {zero, zero, zero, zero}};

  const int arow0 = o0 + lm;        // rows o0..o0+15
  const int arow1 = o0 + 16 + lm;   // rows o0+16..o0+31

#pragma unroll
  for (int c0 = 0; c0 < CIN; c0 += 4) {
    const int ka  = c0 + 2 * half;                       // lane-dependent K pair
    const int kc0 = (ka     < CIN) ? ka     : (CIN - 1); // folds away unless tail
    const int kc1 = (ka + 1 < CIN) ? ka + 1 : (CIN - 1);
    float a0x = W[(size_t)arow0 * CIN + kc0];
    float a0y = W[(size_t)arow0 * CIN + kc1];
    float a1x = W[(size_t)arow1 * CIN + kc0];
    float a1y = W[(size_t)arow1 * CIN + kc1];
    a0x = (ka     < CIN) ? a0x : 0.f;
    a0y = (ka + 1 < CIN) ? a0y : 0.f;
    a1x = (ka     < CIN) ? a1x : 0.f;
    a1y = (ka + 1 < CIN) ? a1y : 0.f;
    v2f A0 = {a0x, a0y};
    v2f A1 = {a1x, a1y};
    const float sc0 = insc[kc0], bi0 = inbi[kc0];
    const float sc1 = insc[kc1], bi1 = inbi[kc1];
#pragma unroll
    for (int t = 0; t < 4; ++t) {
      const int nn = n + t * 16;
      float bx = Xb[(size_t)kc0 * M + nn];
      float by = Xb[(size_t)kc1 * M + nn];
      bx = fmaf(bx, sc0, bi0);
      by = fmaf(by, sc1, bi1);
      if (RELU) { bx = fmaxf(bx, 0.f); by = fmaxf(by, 0.f); }
      bx = (ka     < CIN) ? bx : 0.f;
      by = (ka + 1 < CIN) ? by : 0.f;
      v2f Bv = {bx, by};
      acc[0][t] = __builtin_amdgcn_wmma_f32_16x16x4_f32(false, A0, false, Bv,
                                                        (short)0, acc[0][t], false, false);
      acc[1][t] = __builtin_amdgcn_wmma_f32_16x16x4_f32(false, A1, false, Bv,
                                                        (short)0, acc[1][t], false, false);
    }
  }

  float* Yb = Y + (size_t)b * O * M;
#pragma unroll
  for (int g = 0; g < 2; ++g) {
#pragma unroll
    for (int v = 0; v < 8; ++v) {
      const int o = o0 + 16 * g + v + 8 * half;
      float p = 0.f, p2 = 0.f;
#pragma unroll
      for (int t = 0; t < 4; ++t) {
        float yv = acc[g][t][v];
        Yb[(size_t)o * M + n + t * 16] = yv;
        p  += yv;
        p2 += yv * yv;
      }
      for (int d = 1; d < 16; d <<= 1) {
        p  += __shfl_xor(p,  d, 32);
        p2 += __shfl_xor(p2, d, 32);
      }
      if (lm == 0) {
        atomicAdd(&osum[o],   p);
        atomicAdd(&osumsq[o], p2);
      }
    }
  }
}

// ---------------------------------------------------------------------------
__global__ void bn_finalize_kernel(const float* __restrict__ sum, const float* __restrict__ sumsq,
                                   const float* __restrict__ g, const float* __restrict__ bb,
                                   float* __restrict__ a, float* __restrict__ c2,
                                   int O, float invcnt) {
  int o = blockIdx.x * blockDim.x + threadIdx.x;
  if (o < O) {
    float mean = sum[o] * invcnt;
    float var  = fmaxf(sumsq[o] * invcnt - mean * mean, 0.f);
    float inv  = rsqrtf(var + EPSV);
    float av   = g[o] * inv;
    a[o]  = av;
    c2[o] = bb[o] - mean * av;
  }
}

__global__ void zero_kernel(float* __restrict__ p, int n) {
  int i = blockIdx.x * blockDim.x + threadIdx.x;
  if (i < n) p[i] = 0.f;
}

__global__ void ident_kernel(float* __restrict__ a, float* __restrict__ c2) {
  int i = threadIdx.x;           // <<<1,128>>>
  a[i] = 1.f; c2[i] = 0.f;
}

// ---------------------------------------------------------------------------
// Final BN+ReLU + max-pool over K, write into output feature block.
// ---------------------------------------------------------------------------
__global__ __launch_bounds__(256)
void pool_kernel(const float* __restrict__ Y, const float* __restrict__ a,
                 const float* __restrict__ c2, float* __restrict__ out,
                 int K, int chan_off) {
  int t = blockIdx.x * blockDim.x + threadIdx.x;   // (b, o<128, s)
  int s = t % NPOINT;
  int o = (t / NPOINT) & 127;
  int b = t / (NPOINT * 128);
  const float* y = Y + ((size_t)(b * 128 + o)) * ((size_t)NPOINT * K) + (size_t)s * K;
  float av = a[o], cv = c2[o];
  float m = 0.f;                 // relu(x) >= 0, so max(0, max_k y') is exact
  for (int k = 0; k < K; ++k) m = fmaxf(m, fmaf(y[k], av, cv));
  out[((size_t)b * 256 + chan_off + o) * NPOINT + s] = m;
}

// ---------------------------------------------------------------------------
template <int CIN, int O, int M, int RELU>
static inline void launch_conv(const float* X, const float* W, float* Y,
                               const float* insc, const float* inbi,
                               float* osum, float* osumsq, hipStream_t stream) {
  constexpr int waves = NBATCH * (O / 32) * (M / 64);
  conv_wmma_kernel<CIN, O, M, RELU><<<waves / 8, 256, 0, stream>>>(
      X, W, Y, insc, inbi, osum, osumsq);
}

extern "C" void kernel_launch(void* const* d_in, const int* in_sizes, int n_in,
                              void* d_out, int out_size, void* d_ws, size_t ws_size,
                              hipStream_t stream) {
  const float* xyz  = (const float*)d_in[0];
  const float* feat = (const float*)d_in[1];
  const float* W[2][3]; const float* G[2][3]; const float* Bt[2][3];
  int ii = 2;
  for (int s = 0; s < 2; ++s)
    for (int l = 0; l < 3; ++l) {
      W[s][l]  = (const float*)d_in[ii++];
      G[s][l]  = (const float*)d_in[ii++];
      Bt[s][l] = (const float*)d_in[ii++];
    }

  float* out     = (float*)d_out;
  float* new_xyz = out;                                    // (B, 1024, 3)
  float* outfeat = out + (size_t)NBATCH * NPOINT * 3;      // (B, 256, 1024)

  char* ws = (char*)d_ws;
  const size_t MB = 1ull << 20;
  int*   idxbuf[2] = {(int*)ws, (int*)(ws + 1 * MB)};      // 1MB + 2MB
  float* stats = (float*)(ws + 4 * MB);                    // 3 slots x 512 floats
  float* bufA  = (float*)(ws + 8 * MB);                    // 192MB: up to 96ch x 32K cols
  float* bufB  = (float*)(ws + 8 * MB + 192 * MB);         // 256MB: up to 128ch x 32K cols

  // stats slot layout: [sum(128) | sumsq(128) | a(128) | c2(128)] x 3 slots
  float* slot0 = stats;
  float* slot1 = stats + 512;
  float* ident = stats + 1024;

  fps_kernel<<<NBATCH, 1024, 0, stream>>>(xyz, new_xyz);
  ident_kernel<<<1, 128, 0, stream>>>(ident + 256, ident + 384);

  const float radii[2] = {0.1f, 0.2f};
  const int   ksamp[2] = {16, 32};
  constexpr int M0 = NPOINT * 16;
  constexpr int M1 = NPOINT * 32;

  for (int s = 0; s < 2; ++s) {
    const int K = ksamp[s];
    const int M = NPOINT * K;
    int* idx = idxbuf[s];

    ball_query_kernel<<<(NBATCH * NPOINT * 32) / 256, 256, 0, stream>>>(
        xyz, new_xyz, idx, radii[s] * radii[s], K);
    group_kernel<<<(NBATCH * NPOINT * K) / 256, 256, 0, stream>>>(
        xyz, new_xyz, feat, idx, bufA, K);

    // ---- layer 0: (67 -> 64), identity input transform, no relu ----
    zero_kernel<<<1, 256, 0, stream>>>(slot0, 256);
    if (s == 0) launch_conv<67, 64, M0, 0>(bufA, W[s][0], bufB, ident + 256, ident + 384,
                                           slot0, slot0 + 128, stream);
    else        launch_conv<67, 64, M1, 0>(bufA, W[s][0], bufB, ident + 256, ident + 384,
                                           slot0, slot0 + 128, stream);
    bn_finalize_kernel<<<1, 128, 0, stream>>>(slot0, slot0 + 128, G[s][0], Bt[s][0],
        slot0 + 256, slot0 + 384, 64, 1.0f / (float)((size_t)NBATCH * M));

    // ---- layer 1: (64 -> 64|96), BN+ReLU of layer 0 fused on input ----
    zero_kernel<<<1, 256, 0, stream>>>(slot1, 256);
    if (s == 0) launch_conv<64, 64, M0, 1>(bufB, W[s][1], bufA, slot0 + 256, slot0 + 384,
                                           slot1, slot1 + 128, stream);
    else        launch_conv<64, 96, M1, 1>(bufB, W[s][1], bufA, slot0 + 256, slot0 + 384,
                                           slot1, slot1 + 128, stream);
    bn_finalize_kernel<<<1, 128, 0, stream>>>(slot1, slot1 + 128, G[s][1], Bt[s][1],
        slot1 + 256, slot1 + 384, (s == 0) ? 64 : 96, 1.0f / (float)((size_t)NBATCH * M));

    // ---- layer 2: (64|96 -> 128), BN+ReLU of layer 1 fused on input ----
    zero_kernel<<<1, 256, 0, stream>>>(slot0, 256);
    if (s == 0) launch_conv<64, 128, M0, 1>(bufA, W[s][2], bufB, slot1 + 256, slot1 + 384,
                                            slot0, slot0 + 128, stream);
    else        launch_conv<96, 128, M1, 1>(bufA, W[s][2], bufB, slot1 + 256, slot1 + 384,
                                            slot0, slot0 + 128, stream);
    bn_finalize_kernel<<<1, 128, 0, stream>>>(slot0, slot0 + 128, G[s][2], Bt[s][2],
        slot0 + 256, slot0 + 384, 128, 1.0f / (float)((size_t)NBATCH * M));

    // ---- final BN+ReLU + max-pool over K ----
    pool_kernel<<<(NBATCH * 128 * NPOINT) / 256, 256, 0, stream>>>(
        bufB, slot0 + 256, slot0 + 384, outfeat, K, s * 128);
  }
}